// VoxelTransform_72834055406431
// MI455X (gfx1250) — compile-verified
//
#include <hip/hip_runtime.h>
#include <hip/hip_bf16.h>

#define BATCH   128
#define N_ATOMS 100
#define FEAT    19
#define VOL     32
#define KS      11
#define PADK    5          // KS/2
#define THREADS 512        // 16 wave32s
#define NWAVES  (THREADS / 32)
#define VOL3    (VOL * VOL * VOL)   // 32768

typedef __attribute__((ext_vector_type(2))) float v2f;
typedef __attribute__((ext_vector_type(8))) float v8f;

// One 16x16 fp32 output tile: D = sum_{k=0..31} A(16xk) * B(kx16), via 8x
// V_WMMA_F32_16X16X4_F32.  A/B/D element addresses are generic (row-stride
// based) LDS gathers so the same helper serves all three conv passes.
//
// fp32 WMMA fragment layouts (CDNA5 ISA):
//   A 16x4 : lane L (m = L&15, half = L>>4): v0 = A[m][2*half], v1 = A[m][2*half+1]
//   B 4x16 : lane L (n = L&15, half = L>>4): v0 = B[2*half][n], v1 = B[2*half+1][n]
//   C/D    : VGPR r: M = r + 8*half, N = lane&15
__device__ __forceinline__ void wmma_tile32(const float* __restrict__ A, int rsA,
                                            const float* __restrict__ B, int rsB,
                                            float* __restrict__ D, int rsD,
                                            int lane) {
    const int half = lane >> 4;
    const int q    = lane & 15;
    v8f c = {0.f, 0.f, 0.f, 0.f, 0.f, 0.f, 0.f, 0.f};
#pragma unroll
    for (int k0 = 0; k0 < 32; k0 += 4) {
        const int ka = k0 + 2 * half;
        v2f a, b;
        a.x = A[q * rsA + ka];          // consecutive in k -> ds_load_b64
        a.y = A[q * rsA + ka + 1];
        b.x = B[ka * rsB + q];
        b.y = B[(ka + 1) * rsB + q];
        c = __builtin_amdgcn_wmma_f32_16x16x4_f32(
            /*neg_a=*/false, a, /*neg_b=*/false, b,
            /*c_mod=*/(short)0, c, /*reuse_a=*/false, /*reuse_b=*/false);
    }
#pragma unroll
    for (int r = 0; r < 8; ++r) {
        D[(r + 8 * half) * rsD + q] = c[r];
    }
}

__global__ __launch_bounds__(THREADS) void voxel_gauss_kernel(
    const float* __restrict__ xin,      // (BATCH, N_ATOMS, 3+FEAT)
    const float* __restrict__ kern,     // (FEAT, 1, KS, KS, KS)
    float* __restrict__ out)            // (BATCH, FEAT, VOL, VOL, VOL)
{
    const int f = blockIdx.x;           // feature  [0,19)
    const int b = blockIdx.y;           // batch    [0,128)
    const int tid  = threadIdx.x;
    const int lane = tid & 31;
    const int wave = tid >> 5;

    // ---- LDS: two full 32^3 fp32 volumes (ping-pong) + Toeplitz matrix ----
    __shared__ __align__(16) float bufA[VOL3];      // 128 KB
    __shared__ __align__(16) float bufB[VOL3];      // 128 KB
    __shared__ __align__(16) float tmat[VOL * VOL]; // 4 KB, T[k][x]
    __shared__ float w1[KS];
    __shared__ float axyz[N_ATOMS * 3];
    __shared__ float afeat[N_ATOMS];
    __shared__ int   amask[N_ATOMS];
    __shared__ float bounds[6];         // minx,miny,minz,maxx,maxy,maxz

    // ---- Phase 1: load atoms; reduce 3D kernel -> normalized 1D weights ----
    if (tid < N_ATOMS) {
        const float* ap = xin + ((size_t)b * N_ATOMS + tid) * (3 + FEAT);
        float px = ap[0], py = ap[1], pz = ap[2];
        axyz[tid * 3 + 0] = px;
        axyz[tid * 3 + 1] = py;
        axyz[tid * 3 + 2] = pz;
        afeat[tid] = ap[3 + f];
        amask[tid] = (px != 0.f) & (py != 0.f) & (pz != 0.f);
    }
    if (tid < KS) {
        // w1[t] = sum_{j,k} kernel[f,0,t,j,k]  (exact 1D factor of the
        // normalized separable kernel)
        const float* kp = kern + (size_t)f * (KS * KS * KS) + tid * (KS * KS);
        float s = 0.f;
        for (int j = 0; j < KS * KS; ++j) s += kp[j];
        w1[tid] = s;
    }
    __syncthreads();

    // ---- Phase 2: zero volume, build Toeplitz T, reduce bounds ----
    {
        float4 z4 = make_float4(0.f, 0.f, 0.f, 0.f);
        float4* pA = (float4*)bufA;
#pragma unroll
        for (int i = tid; i < VOL3 / 4; i += THREADS) pA[i] = z4;
    }
    if (tid < VOL * VOL) {
        int i = tid >> 5, j = tid & 31;
        int d = i - j;
        int ad = d < 0 ? -d : d;
        tmat[tid] = (ad <= PADK) ? w1[ad + PADK] : 0.f;
    }
    if (tid < 6) {
        const int axis = tid % 3;
        const bool mx  = tid >= 3;
        float r = mx ? -__builtin_inff() : __builtin_inff();
        for (int a = 0; a < N_ATOMS; ++a) {
            if (amask[a]) {
                float v = axyz[a * 3 + axis];
                r = mx ? fmaxf(r, v) : fminf(r, v);
            }
        }
        bounds[tid] = r;
    }
    __syncthreads();

    // ---- Phase 3: voxelize (LDS float atomics: ds_add_f32) ----
    if (tid < N_ATOMS && amask[tid]) {
        const float fv = afeat[tid];
        int c[3];
#pragma unroll
        for (int ax = 0; ax < 3; ++ax) {
            float mn = bounds[ax], mxv = bounds[3 + ax];
            float t = (axyz[tid * 3 + ax] - mn) / (mxv - mn) * (float)(VOL - 1);
            c[ax] = (int)floorf(t);
        }
#pragma unroll
        for (int dz = -1; dz <= 1; ++dz)
#pragma unroll
            for (int dy = -1; dy <= 1; ++dy)
#pragma unroll
                for (int dx = -1; dx <= 1; ++dx) {
                    int tx = c[0] + dx, ty = c[1] + dy, tz = c[2] + dz;
                    if (((unsigned)tx < VOL) & ((unsigned)ty < VOL) & ((unsigned)tz < VOL)) {
                        atomicAdd(&bufA[(tz * VOL + ty) * VOL + tx], fv);
                    }
                }
    }
    __syncthreads();

    // ---- Phase 4: separable conv as three WMMA matmul passes ----
    // Pass X: bufB[(z,y),x] = sum_k bufA[(z,y),k] * T[k,x]
    for (int t = wave; t < 128; t += NWAVES) {
        const int rowT = t >> 1;        // 0..63  (16-row groups over (z,y))
        const int colT = t & 1;         // 0..1
        wmma_tile32(bufA + rowT * 16 * VOL, VOL,
                    tmat + colT * 16,     VOL,
                    bufB + rowT * 16 * VOL + colT * 16, VOL, lane);
    }
    __syncthreads();

    // Pass Y: bufA[z,y,x] = sum_k T[y,k] * bufB[z,k,x]   (T symmetric)
    for (int t = wave; t < 128; t += NWAVES) {
        const int z  = t >> 2;          // 0..31
        const int yT = (t >> 1) & 1;    // 0..1
        const int xT = t & 1;           // 0..1
        wmma_tile32(tmat + yT * 16 * VOL, VOL,
                    bufB + z * (VOL * VOL) + xT * 16, VOL,
                    bufA + z * (VOL * VOL) + yT * 16 * VOL + xT * 16, VOL, lane);
    }
    __syncthreads();

    // Pass Z: bufB[z,(y,x)] = sum_k T[z,k] * bufA[k,(y,x)]
    for (int t = wave; t < 128; t += NWAVES) {
        const int zT   = t >> 6;        // 0..1
        const int colT = t & 63;        // 0..63 over (y,x)
        wmma_tile32(tmat + zT * 16 * VOL, VOL,
                    bufA + colT * 16,    VOL * VOL,
                    bufB + zT * 16 * (VOL * VOL) + colT * 16, VOL * VOL, lane);
    }
    __syncthreads();

    // ---- Phase 5: coalesced 128-bit stores of the finished volume ----
    {
        float4* dst = (float4*)(out + (((size_t)b * FEAT + f) << 15));
        const float4* src = (const float4*)bufB;
#pragma unroll
        for (int i = tid; i < VOL3 / 4; i += THREADS) dst[i] = src[i];
    }
}

extern "C" void kernel_launch(void* const* d_in, const int* in_sizes, int n_in,
                              void* d_out, int out_size, void* d_ws, size_t ws_size,
                              hipStream_t stream) {
    (void)in_sizes; (void)n_in; (void)d_ws; (void)ws_size; (void)out_size;
    const float* x    = (const float*)d_in[0];
    const float* kern = (const float*)d_in[1];
    float* out        = (float*)d_out;

    dim3 grid(FEAT, BATCH, 1);   // one workgroup per (batch, feature) volume
    dim3 block(THREADS, 1, 1);
    voxel_gauss_kernel<<<grid, block, 0, stream>>>(x, kern, out);
}